// NeighborSample_33930241638702
// MI455X (gfx1250) — compile-verified
//
#include <hip/hip_runtime.h>
#include <stdint.h>

// ---------------------------------------------------------------------------
// NeighborSample (5x5 zero-padded im2col), MI455X / gfx1250.
// 24 MiB in, 600 MiB out -> pure data movement, store-BW bound (~27us floor).
//   * one workgroup per (b,y) pixel row; stage rows y-2..y+2 of all 192
//     channels into ~322KB of WGP LDS
//   * staging via the CDNA5 Tensor Data Mover: one TENSOR_LOAD_TO_LDS
//     descriptor per window row (2D tile: 64 floats x 192 channels,
//     channel stride 4096), tracked by TENSORcnt      [CDNA5 flagship path]
//   * TDM pad feature gives each channel an 84-float LDS pitch and regions
//     are staggered by 4 floats -> emit-phase bank index 4*(5c+i)+x+j is
//     strictly increasing across a wave => near-conflict-free ds_load_b32
//   * output emitted as 16B-aligned nontemporal b128 stores (write-once,
//     keeps L2 for the 25x-reused input)
//   * fallback (builtin absent): per-lane global_load_async_to_lds_b128
//     into the identical layout
// ---------------------------------------------------------------------------

typedef __attribute__((ext_vector_type(4))) float        v4f;
typedef __attribute__((ext_vector_type(4))) unsigned int u32x4;
typedef __attribute__((ext_vector_type(8))) int          i32x8;
typedef __attribute__((ext_vector_type(4))) int          i32x4;

#define NS_B   8
#define NS_C   192
#define NS_H   64
#define NS_W   64
#define NS_K   5
#define NS_PAD 2

#define CH_STRIDE 84                   // floats per channel line in LDS (64 + 20 TDM pad)
#define R_REGION  16128                // floats written per window-row region (192*84-20 rounded up)
#define R_STRIDE  16132                // region pitch: +4 float stagger (keeps 16B align)
#define LDS_FLOATS (4 * R_STRIDE + R_REGION)   // 80656 floats = 322624 B  (< 320KB WGP LDS)

#if defined(__has_builtin)
#  if __has_builtin(__builtin_amdgcn_tensor_load_to_lds)
#    define HAVE_TDM 1
#  endif
#endif
#ifndef HAVE_TDM
#  define HAVE_TDM 0
#endif

// Generic -> LDS(addrspace 3) byte offset (wave-relative).
__device__ __forceinline__ uint32_t lds_byte_offset(const void* p) {
  return (uint32_t)(uintptr_t)(__attribute__((address_space(3))) const void*)p;
}

__device__ __forceinline__ void wait_tensorcnt0() {
#if defined(__has_builtin) && __has_builtin(__builtin_amdgcn_s_wait_tensorcnt)
  __builtin_amdgcn_s_wait_tensorcnt(0);
#else
  asm volatile("s_wait_tensorcnt 0" ::: "memory");
#endif
}

#if !HAVE_TDM
__device__ __forceinline__ void async_copy_b128(uint32_t lds_byte, const float* gaddr) {
  asm volatile("global_load_async_to_lds_b128 %0, %1, off"
               :: "v"(lds_byte), "v"(gaddr) : "memory");
}
__device__ __forceinline__ void wait_asynccnt0() {
  asm volatile("s_wait_asynccnt 0" ::: "memory");
}
#endif

__global__ __launch_bounds__(1024)
void NeighborSample_gfx1250_kernel(const float* __restrict__ in,
                                   float* __restrict__ out) {
  extern __shared__ __align__(16) float smem[];

  const int tid   = (int)threadIdx.x;
  const int rowid = (int)blockIdx.x;          // 0 .. B*H-1 (512 blocks)
  const int bi    = rowid >> 6;
  const int y     = rowid & (NS_H - 1);

  const float*   in_b      = in + (size_t)bi * NS_C * NS_H * NS_W;
  const uint32_t smem_base = lds_byte_offset(smem);

  // ---- Zero-fill regions of window rows that fall outside the image ------
#pragma unroll
  for (int r = 0; r < NS_K; ++r) {
    int yin = y + r - NS_PAD;
    if ((unsigned)yin >= (unsigned)NS_H) {    // uniform per block
      v4f z = {0.f, 0.f, 0.f, 0.f};
#pragma unroll
      for (int it = 0; it < 4; ++it) {        // 16128/4 = 4032 float4
        int t = it * 1024 + tid;
        if (t < R_REGION / 4) *(v4f*)(&smem[r * R_STRIDE + t * 4]) = z;
      }
    }
  }

#if HAVE_TDM
  // ---- Phase 1: TDM descriptors, one per valid window row (wave 0 only) --
  if (tid < 32) {
#pragma unroll
    for (int r = 0; r < NS_K; ++r) {
      int yin = y + r - NS_PAD;
      if ((unsigned)yin < (unsigned)NS_H) {
        uint64_t ga = (uint64_t)(uintptr_t)(in_b + (size_t)yin * NS_W);
        u32x4 g0 = {0u, 0u, 0u, 0u};
        g0[0] = 1u;                                        // count = 1 valid D#
        g0[1] = smem_base + (uint32_t)(r * R_STRIDE * 4);  // lds_addr (bytes)
        g0[2] = (uint32_t)ga;                              // global_addr[31:0]
        g0[3] = (uint32_t)((ga >> 32) & 0x01FFFFFFu)       // global_addr[56:32]
              | (2u << 30);                                // type = 2 (image)
        i32x8 g1 = {0, 0, 0, 0, 0, 0, 0, 0};
        g1[0] = (int)((2u << 16)      // data_size = 4B
                    | (1u << 20)      // pad_enable
                    | (5u << 22)      // pad_interval: every 64 DWORDs
                    | (19u << 25));   // pad_amount: 20 DWORDs -> 84f channel pitch
        g1[1] = (int)((unsigned)NS_W << 16);        // tensor_dim0 = 64
        g1[2] = (int)((unsigned)NS_C << 16);        // tensor_dim1 = 192
        g1[3] = (int)((unsigned)NS_W << 16);        // tile_dim0 = 64
        g1[4] = NS_C;                               // tile_dim1 = 192 (tile_dim2 = 0)
        g1[5] = NS_H * NS_W;                        // tensor_dim0_stride = 4096 (channel)
        i32x4 gz = {0, 0, 0, 0};                    // 2D: groups 2/3 unused
#if __clang_major__ >= 23
        i32x8 gz8 = {0, 0, 0, 0, 0, 0, 0, 0};
        __builtin_amdgcn_tensor_load_to_lds(g0, g1, gz, gz, gz8, 0);
#else
        __builtin_amdgcn_tensor_load_to_lds(g0, g1, gz, gz, 0);
#endif
      }
    }
    wait_tensorcnt0();        // issuer wave drains TENSORcnt before barrier
  }
#else
  // ---- Phase 1 fallback: per-lane async copies into the same layout ------
#pragma unroll
  for (int it = 0; it < 15; ++it) {
    int k    = it * 1024 + tid;          // 192*5*16 = 15360 16B chunks
    int c    = k / 80;
    int rr   = k - c * 80;
    int r    = rr >> 4;
    int xq   = rr & 15;
    int yin  = y + r - NS_PAD;
    if ((unsigned)yin < (unsigned)NS_H) {
      const float* g = in_b + ((size_t)c * NS_H + (size_t)yin) * NS_W + xq * 4;
      int fidx = r * R_STRIDE + c * CH_STRIDE + xq * 4;   // 16B aligned
      async_copy_b128(smem_base + (uint32_t)(fidx * 4), g);
    }
  }
  wait_asynccnt0();
#endif
  __syncthreads();

  // ---- Phase 2: 64 px * 192 ch * 25 taps = 76800 float4 NT stores --------
  // out[(row,x), c, i, j] = lds[i*R_STRIDE + c*CH_STRIDE + (x+j-2)]
  float* out_row = out + (size_t)rowid * (NS_W * NS_C * NS_K * NS_K);  // 307200
#pragma unroll 1
  for (int it = 0; it < 75; ++it) {
    int idx4 = it * 1024 + tid;          // 0..76799; 1200 float4 per pixel
    int x    = idx4 / 1200;
    int f0   = (idx4 - x * 1200) * 4;    // flat [c][i][j] offset
    v4f v;
#pragma unroll
    for (int e = 0; e < 4; ++e) {
      int fe  = f0 + e;
      int q   = fe / 5;                  // q = 5c + i
      int j   = fe - q * 5;
      int c   = q / 5;
      int i   = q - c * 5;
      int xin = x + j - NS_PAD;
      float val = 0.f;
      if ((unsigned)xin < (unsigned)NS_W)
        val = smem[i * R_STRIDE + c * CH_STRIDE + xin];
      v[e] = val;
    }
    __builtin_nontemporal_store(v, (v4f*)(out_row + (size_t)idx4 * 4));
  }
}

extern "C" void kernel_launch(void* const* d_in, const int* in_sizes, int n_in,
                              void* d_out, int out_size, void* d_ws, size_t ws_size,
                              hipStream_t stream) {
  (void)in_sizes; (void)n_in; (void)out_size; (void)d_ws; (void)ws_size;
  const float* in  = (const float*)d_in[0];
  float*       out = (float*)d_out;
  dim3 grid(NS_B * NS_H);                // 512 workgroups, one per (b, y)
  dim3 block(1024);                      // 32 wave32
  size_t shmem = (size_t)LDS_FLOATS * sizeof(float);   // 322624 B dynamic LDS
  hipLaunchKernelGGL(NeighborSample_gfx1250_kernel, grid, block, shmem, stream,
                     in, out);
}